// TimeVaryingBiquad_56264071577775
// MI455X (gfx1250) — compile-verified
//
#include <hip/hip_runtime.h>
#include <stdint.h>

// ---------------- problem geometry ----------------
#define TLEN   131072           // samples per row
#define BATCH  32               // rows
#define L      64               // chunk length (samples per lane)
#define CPW    32               // chunks per wave (one per lane)
#define TILE   (CPW * L)        // 2048 samples staged per wave
#define TPR    (TLEN / TILE)    // 64 tiles per row
#define NC     (TLEN / L)       // 2048 chunks per row
#define TOTCH  (BATCH * NC)     // 65536 chunks total
#define LDSROW (L + 1)          // padded chunk pitch in LDS (bank-conflict free)
#define CPB    (NC / 32)        // 64 chunks per lane in the K2 row-scan

typedef uint32_t u32;
typedef __attribute__((ext_vector_type(4))) u32 u32x4;
typedef __attribute__((ext_vector_type(8))) u32 u32x8;

// ---------------- TDM: stage a 32x64 fp32 tile (contiguous in global) into
// padded LDS (1 dword pad after every 64 dwords) ----------------
// D# per CDNA5 ISA ch.8: group0 (4 SGPR) + group1 (8 SGPR), groups 2/3 = NULL (2D).
__device__ __forceinline__ void tdm_load_tile(u32 lds_byte, const float* gsrc) {
  uint64_t ga = (uint64_t)(uintptr_t)gsrc;
  u32x4 g0;
  g0[0] = 1u;                                             // count=1, user descriptor
  g0[1] = lds_byte;                                       // lds_addr (bytes)
  g0[2] = (u32)ga;                                        // global_addr[31:0]
  g0[3] = ((u32)(ga >> 32) & 0x01FFFFFFu) | (2u << 30);   // global_addr[56:32] | type=2
  u32x8 g1;
  // data_size=2 (4B), pad_enable=1, pad_interval=5 (every 64 dwords), pad_amount=0 (1 dword)
  g1[0] = (2u << 16) | (1u << 20) | (5u << 22);
  g1[1] = ((u32)L & 0xFFFFu) << 16;                       // tensor_dim0 = 64 (bits 79:48 lo)
  g1[2] = ((u32)CPW) << 16;                               // tensor_dim0 hi=0 | tensor_dim1 = 32
  g1[3] = ((u32)L) << 16;                                 // tensor_dim1 hi=0 | tile_dim0 = 64
  g1[4] = (u32)CPW;                                       // tile_dim1 = 32, tile_dim2 = 0
  g1[5] = (u32)L;                                         // tensor_dim0_stride = 64 (lo32)
  g1[6] = 0u;
  g1[7] = 0u;
  asm volatile("tensor_load_to_lds %0, %1" :: "s"(g0), "s"(g1) : "memory");
}

__device__ __forceinline__ void tdm_wait0() {
  __builtin_amdgcn_s_wait_tensorcnt(0);
  asm volatile("" ::: "memory");
}

// ---------------- biquad coefficient math (per sample) ----------------
__device__ __forceinline__ void coeffs(float c, float r,
                                       float& a1, float& a2,
                                       float& invA0, float& cw) {
  const float MINW = 0.001f;
  const float WSC  = 3.1395926535897933f;   // (pi-0.001) - 0.001
  const float LQ0  = -0.34658318f;          // ln(0.7071)
  const float LQS  = 1.73287754f;           // ln(4) - ln(0.7071)
  float w  = fmaf(c, WSC, MINW);
  float q  = __expf(fmaf(r, LQS, LQ0));
  float sw = __sinf(w);
  cw       = __cosf(w);
  float alpha = 0.5f * sw * __builtin_amdgcn_rcpf(q);
  invA0    = __builtin_amdgcn_rcpf(1.0f + alpha);
  a1       = (-2.0f * cw) * invA0;
  a2       = (1.0f - alpha) * invA0;
}

// ---------------- K1: per-chunk summaries (2x2 transition M, offset v) ----------------
__global__ __launch_bounds__(32) void biquad_k1_summary(
    const float* __restrict__ x, const float* __restrict__ cut,
    const float* __restrict__ res, float* __restrict__ sum) {
  __shared__ float sX[CPW * LDSROW];
  __shared__ float sC[CPW * LDSROW];
  __shared__ float sR[CPW * LDSROW];
  const int lane = threadIdx.x;                 // 0..31, one chunk per lane
  const int gt   = blockIdx.x;                  // tile id
  const int row  = gt / TPR;
  const int tir  = gt - row * TPR;
  const size_t base = (size_t)row * TLEN + (size_t)tir * TILE;

  tdm_load_tile((u32)(uintptr_t)sX, x   + base);
  tdm_load_tile((u32)(uintptr_t)sC, cut + base);
  tdm_load_tile((u32)(uintptr_t)sR, res + base);
  tdm_wait0();

  const float* cx = sX + lane * LDSROW;
  const float* cc = sC + lane * LDSROW;
  const float* cr = sR + lane * LDSROW;

  float y1 = 0.f, y2 = 0.f;       // particular solution (zero init)
  float p11 = 1.f, p12 = 0.f;     // homogeneous: dependence on y[-1]
  float p21 = 0.f, p22 = 1.f;     // homogeneous: dependence on y[-2]
  #pragma unroll 4
  for (int t = 0; t < L; ++t) {
    float a1, a2, ia0, cw;
    coeffs(cc[t], cr[t], a1, a2, ia0, cw);
    float yn  = fmaf(-a1, y1,  fmaf(-a2, y2,  cx[t]));
    float p1n = fmaf(-a1, p11, -a2 * p12);
    float p2n = fmaf(-a1, p21, -a2 * p22);
    y2 = y1; y1 = yn;
    p12 = p11; p11 = p1n;
    p22 = p21; p21 = p2n;
  }
  const int chunk = row * NC + tir * CPW + lane;   // SoA planes -> coalesced stores
  sum[0 * TOTCH + chunk] = p11;   // m11
  sum[1 * TOTCH + chunk] = p21;   // m12
  sum[2 * TOTCH + chunk] = p12;   // m21
  sum[3 * TOTCH + chunk] = p22;   // m22
  sum[4 * TOTCH + chunk] = y1;    // v1
  sum[5 * TOTCH + chunk] = y2;    // v2
}

// ---------------- K2: per-row wave scan of chunk summaries ----------------
// One 32-lane wave per row. Phase A: lane composes its 64 chunk affine maps.
// Phase B: Hillis-Steele affine scan across lanes via __shfl_up (wave32).
// Phase C: lane replays its 64 chunks from its entry state, emitting per-chunk
// initial conditions (y[start-1], y[start-2]).
__global__ __launch_bounds__(32) void biquad_k2_scan(
    const float* __restrict__ sum, float* __restrict__ ini) {
  const int row  = blockIdx.x;     // 0..31
  const int lane = threadIdx.x;    // 0..31
  const int off  = row * NC + lane * CPB;

  // Phase A: compose (M,v) over this lane's 64 chunks: new = Mc*old, Mc*v + vc
  float m11 = 1.f, m12 = 0.f, m21 = 0.f, m22 = 1.f, v1 = 0.f, v2 = 0.f;
  for (int j = 0; j < CPB; ++j) {
    const int k = off + j;
    float c11 = sum[k],             c12 = sum[TOTCH + k];
    float c21 = sum[2 * TOTCH + k], c22 = sum[3 * TOTCH + k];
    float w1  = sum[4 * TOTCH + k], w2  = sum[5 * TOTCH + k];
    float n11 = fmaf(c11, m11, c12 * m21);
    float n12 = fmaf(c11, m12, c12 * m22);
    float n21 = fmaf(c21, m11, c22 * m21);
    float n22 = fmaf(c21, m12, c22 * m22);
    float nv1 = fmaf(c11, v1, fmaf(c12, v2, w1));
    float nv2 = fmaf(c21, v1, fmaf(c22, v2, w2));
    m11 = n11; m12 = n12; m21 = n21; m22 = n22; v1 = nv1; v2 = nv2;
  }

  // Phase B: inclusive affine scan across lanes (cur after prev): cur ∘ prev
  #pragma unroll
  for (int d = 1; d < 32; d <<= 1) {
    float p11 = __shfl_up(m11, d), p12 = __shfl_up(m12, d);
    float p21 = __shfl_up(m21, d), p22 = __shfl_up(m22, d);
    float pv1 = __shfl_up(v1, d),  pv2 = __shfl_up(v2, d);
    if (lane >= d) {
      float n11 = fmaf(m11, p11, m12 * p21);
      float n12 = fmaf(m11, p12, m12 * p22);
      float n21 = fmaf(m21, p11, m22 * p21);
      float n22 = fmaf(m21, p12, m22 * p22);
      float nv1 = fmaf(m11, pv1, fmaf(m12, pv2, v1));
      float nv2 = fmaf(m21, pv1, fmaf(m22, pv2, v2));
      m11 = n11; m12 = n12; m21 = n21; m22 = n22; v1 = nv1; v2 = nv2;
    }
  }
  // Exclusive: entry state for this lane = inclusive v of lane-1 (zero init row)
  float s1 = __shfl_up(v1, 1);
  float s2 = __shfl_up(v2, 1);
  if (lane == 0) { s1 = 0.f; s2 = 0.f; }

  // Phase C: replay this lane's chunks, writing entry states per chunk
  for (int j = 0; j < CPB; ++j) {
    const int k = off + j;
    ini[k]         = s1;
    ini[TOTCH + k] = s2;
    float c11 = sum[k],             c12 = sum[TOTCH + k];
    float c21 = sum[2 * TOTCH + k], c22 = sum[3 * TOTCH + k];
    float w1  = sum[4 * TOTCH + k], w2  = sum[5 * TOTCH + k];
    float n1 = fmaf(c11, s1, fmaf(c12, s2, w1));
    float n2 = fmaf(c21, s1, fmaf(c22, s2, w2));
    s1 = n1; s2 = n2;
  }
}

// ---------------- K3: re-run with true init state, fused FIR, coalesced out ----------------
__global__ __launch_bounds__(32) void biquad_k3_apply(
    const float* __restrict__ x, const float* __restrict__ cut,
    const float* __restrict__ res, const float* __restrict__ ini,
    float* __restrict__ out) {
  __shared__ float sX[CPW * LDSROW];
  __shared__ float sC[CPW * LDSROW];
  __shared__ float sR[CPW * LDSROW];
  __shared__ float sO[CPW * LDSROW];
  const int lane = threadIdx.x;
  const int gt   = blockIdx.x;
  const int row  = gt / TPR;
  const int tir  = gt - row * TPR;
  const size_t base = (size_t)row * TLEN + (size_t)tir * TILE;

  tdm_load_tile((u32)(uintptr_t)sX, x   + base);
  tdm_load_tile((u32)(uintptr_t)sC, cut + base);
  tdm_load_tile((u32)(uintptr_t)sR, res + base);

  const int chunk = row * NC + tir * CPW + lane;
  float y1 = ini[chunk];            // y[start-1]
  float y2 = ini[TOTCH + chunk];    // y[start-2]

  tdm_wait0();

  const float* cx = sX + lane * LDSROW;
  const float* cc = sC + lane * LDSROW;
  const float* cr = sR + lane * LDSROW;
  float*       co = sO + lane * LDSROW;

  #pragma unroll 4
  for (int t = 0; t < L; ++t) {
    float a1, a2, ia0, cw;
    coeffs(cc[t], cr[t], a1, a2, ia0, cw);
    float b1v = (1.0f - cw) * ia0;
    float b0v = 0.5f * b1v;                       // b2 == b0 (lowpass)
    float yn  = fmaf(-a1, y1, fmaf(-a2, y2, cx[t]));
    // out[t] = b0*y[t-2] + b1*y[t-1] + b0*y[t]
    co[t] = fmaf(b0v, y2, fmaf(b1v, y1, b0v * yn));
    y2 = y1; y1 = yn;
  }
  __syncthreads();                                // order LDS writes vs cross-lane reads
  // padded LDS -> linear global, fully coalesced
  float* op = out + base;
  #pragma unroll 4
  for (int j = 0; j < L; ++j) {
    const int k  = j * 32 + lane;                 // 0..2047 contiguous
    const int c  = k >> 6;
    const int t0 = k & (L - 1);
    op[k] = sO[c * LDSROW + t0];
  }
}

// ---------------- host-side launch ----------------
extern "C" void kernel_launch(void* const* d_in, const int* in_sizes, int n_in,
                              void* d_out, int out_size, void* d_ws, size_t ws_size,
                              hipStream_t stream) {
  const float* x   = (const float*)d_in[0];
  const float* cut = (const float*)d_in[1];
  const float* res = (const float*)d_in[2];
  float* out = (float*)d_out;

  float* sum = (float*)d_ws;              // 6 planes of TOTCH floats
  float* ini = sum + (size_t)6 * TOTCH;   // 2 planes of TOTCH floats (2 MB total ws)

  const int tiles = BATCH * TPR;          // 2048 single-wave blocks
  biquad_k1_summary<<<tiles, 32, 0, stream>>>(x, cut, res, sum);
  biquad_k2_scan   <<<BATCH, 32, 0, stream>>>(sum, ini);
  biquad_k3_apply  <<<tiles, 32, 0, stream>>>(x, cut, res, ini, out);
}